// CustomLoss_46342697124317
// MI455X (gfx1250) — compile-verified
//
#include <hip/hip_runtime.h>
#include <stdint.h>

typedef float v2f __attribute__((ext_vector_type(2)));
typedef float v8f __attribute__((ext_vector_type(8)));

#define C_DIM 1024
#define TPB   256

// Full 32-lane sum using V_WMMA_F32_16X16X4_F32 with B = ones.
// A layout (32-bit 16x4): lanes 0-15 {VGPR0=K0, VGPR1=K1}, lanes 16-31 {VGPR0=K2, VGPR1=K3}.
// Put s in VGPR0, 0 in VGPR1  =>  D[m][n] = s_m + s_{m+16}.
// Lane n<16 holds D[0..7][n] in d[0..7]; lane n>=16 holds D[8..15][n-16].
// sum(d[0..7]) + shfl_xor(.,16) = total over all 32 lanes, in every lane.
__device__ __forceinline__ float wave_sum32_wmma(float s) {
  v2f a; a[0] = s;    a[1] = 0.0f;
  v2f b; b[0] = 1.0f; b[1] = 1.0f;
  v8f c = {};
  v8f d = __builtin_amdgcn_wmma_f32_16x16x4_f32(false, a, false, b,
                                                (short)0, c, false, false);
  float t = ((d[0] + d[1]) + (d[2] + d[3])) + ((d[4] + d[5]) + (d[6] + d[7]));
  t += __shfl_xor(t, 16, 32);
  return t;
}

__global__ void __launch_bounds__(TPB)
row_loss_kernel(const float* __restrict__ outputs,
                const int*   __restrict__ targets,
                const int*   __restrict__ ages,
                const float* __restrict__ weight,
                float*       __restrict__ per_row)
{
  __shared__ __align__(16) float row[C_DIM];
  __shared__ float redmax[8];
  __shared__ float redsum[8];

  const int r    = blockIdx.x;
  const int tid  = threadIdx.x;
  const int wid  = tid >> 5;
  const int lane = tid & 31;

  // ---- Stage the row global -> LDS via the CDNA5 async path (B128/lane) ----
  const float* base = outputs + (size_t)r * C_DIM;
  uint32_t loff  = (uint32_t)(uintptr_t)(&row[0]) + (uint32_t)(tid * 16);
  uint64_t gaddr = (uint64_t)(uintptr_t)base + (uint64_t)(tid * 16);
  asm volatile("global_load_async_to_lds_b128 %0, %1, off th:TH_LOAD_NT"
               :: "v"(loff), "v"(gaddr) : "memory");
  asm volatile("s_wait_asynccnt 0" ::: "memory");
  __syncthreads();

  // ---- Each thread owns 4 contiguous logits; work in y = x / T (T = 2) ----
  const float4 v = *reinterpret_cast<const float4*>(&row[tid * 4]);
  const float y0 = v.x * 0.5f, y1 = v.y * 0.5f, y2 = v.z * 0.5f, y3 = v.w * 0.5f;

  // ---- Block max ----
  float m = fmaxf(fmaxf(y0, y1), fmaxf(y2, y3));
#pragma unroll
  for (int off = 16; off > 0; off >>= 1)
    m = fmaxf(m, __shfl_xor(m, off, 32));
  if (lane == 0) redmax[wid] = m;
  __syncthreads();
  float bm = redmax[0];
#pragma unroll
  for (int w = 1; w < 8; ++w) bm = fmaxf(bm, redmax[w]);

  // ---- Block sum of exp(y - max): v_exp_f32 then WMMA wave-sum ----
  const float L2E = 1.4426950408889634f;
  float e = __builtin_amdgcn_exp2f((y0 - bm) * L2E)
          + __builtin_amdgcn_exp2f((y1 - bm) * L2E)
          + __builtin_amdgcn_exp2f((y2 - bm) * L2E)
          + __builtin_amdgcn_exp2f((y3 - bm) * L2E);
  float ws = wave_sum32_wmma(e);
  if (lane == 0) redsum[wid] = ws;
  __syncthreads();

  // ---- Per-row loss (only needs logp at t and t+1; read logits from LDS) ----
  if (tid == 0) {
    float bsum = 0.0f;
#pragma unroll
    for (int w = 0; w < 8; ++w) bsum += redsum[w];
    const float LN2 = 0.6931471805599453f;
    const float lse = bm + __builtin_amdgcn_logf(bsum) * LN2;   // log-sum-exp of y

    const int   t     = targets[r];
    const float age   = (float)ages[r];
    const float delta = (age > 50.0f && age < 60.0f) ? (age - 50.0f) * 0.1f : 0.0f;

    const float yt  = row[t]     * 0.5f;
    const float yt1 = row[t + 1] * 0.5f;
    const float loss = -((1.0f - delta) * (yt  - lse) * weight[t]
                       +  delta         * (yt1 - lse) * weight[t + 1]);
    per_row[r] = loss;
  }
}

// Deterministic mean over n per-row losses: fixed strided order per thread,
// WMMA wave-sum, fixed 8-wave combine. Same result on every invocation.
__global__ void __launch_bounds__(TPB)
reduce_mean_kernel(const float* __restrict__ pr, float* __restrict__ out, int n)
{
  __shared__ float red[8];
  const int tid  = threadIdx.x;
  const int wid  = tid >> 5;
  const int lane = tid & 31;

  float s = 0.0f;
  for (int i = tid; i < n; i += TPB) s += pr[i];

  float t = wave_sum32_wmma(s);
  if (lane == 0) red[wid] = t;
  __syncthreads();

  if (tid == 0) {
    float tot = 0.0f;
#pragma unroll
    for (int w = 0; w < 8; ++w) tot += red[w];
    out[0] = tot * (1.0f / (float)n);
  }
}

extern "C" void kernel_launch(void* const* d_in, const int* in_sizes, int n_in,
                              void* d_out, int out_size, void* d_ws, size_t ws_size,
                              hipStream_t stream)
{
  const float* outputs = (const float*)d_in[0];   // (B, C) f32
  const int*   targets = (const int*)  d_in[1];   // (B,)   i32
  const int*   ages    = (const int*)  d_in[2];   // (B,)   i32
  const float* weight  = (const float*)d_in[3];   // (C,)   f32
  const int B = in_sizes[1];                      // 32768

  float* per_row = (float*)d_ws;                  // B floats of scratch

  row_loss_kernel<<<B, TPB, 0, stream>>>(outputs, targets, ages, weight, per_row);
  reduce_mean_kernel<<<1, TPB, 0, stream>>>(per_row, (float*)d_out, B);
}